// Decoder_29403346108474
// MI455X (gfx1250) — compile-verified
//
#include <hip/hip_runtime.h>
#include <hip/hip_bf16.h>
#include <math.h>

// Problem constants (from reference): V=32000, E=1024, H=1024, A=10, S=512, B=64
#define CB   64
#define CH   1024
#define CE   1024
#define CS   512
#define CV   32000
#define CA   10

typedef float v2f __attribute__((ext_vector_type(2)));
typedef float v8f __attribute__((ext_vector_type(8)));

// D = A(16x4 f32) * B(4x16 f32) + C(16x16 f32)  -> v_wmma_f32_16x16x4_f32
__device__ __forceinline__ v8f wmma_f32_k4(v2f a, v2f b, v8f c) {
  return __builtin_amdgcn_wmma_f32_16x16x4_f32(
      /*neg_a=*/false, a, /*neg_b=*/false, b,
      /*c_mod=*/(short)0, c, /*reuse_a=*/false, /*reuse_b=*/false);
}

// ---------------------------------------------------------------------------
// K0a: zero-padded attention weight slab: Wa1p[r][k] = (r<A) ? Wa1[r][H+k] : 0
// (16 x H). Removes all conditional loads from the WMMA attention loop.
// grid = 16*H/256 blocks, block = 256
// ---------------------------------------------------------------------------
__global__ __launch_bounds__(256) void pad_wa1_kernel(
    const float* __restrict__ Wa1, float* __restrict__ Wa1p) {
  const int idx = blockIdx.x * 256 + threadIdx.x;   // 0 .. 16*CH-1
  const int row = idx >> 10;                        // /CH
  const int col = idx & (CH - 1);
  Wa1p[idx] = (row < CA) ? Wa1[row * (2 * CH) + CH + col] : 0.f;
}

// ---------------------------------------------------------------------------
// K0b: hproj[b][a] = dot(Wa1[a, 0:H], h_last[b, :])  (padded to 16 cols, zero)
// grid = (16, B), block = 32 (one wave)
// ---------------------------------------------------------------------------
__global__ __launch_bounds__(32) void hproj_kernel(
    const float* __restrict__ Wa1, const float* __restrict__ hlast,
    float* __restrict__ hproj) {
  const int a = blockIdx.x;
  const int b = blockIdx.y;
  const int lane = threadIdx.x;
  float s = 0.f;
  if (a < CA) {
    const float* wrow = Wa1 + a * (2 * CH);
    const float* h = hlast + b * CH;
    for (int i = lane; i < CH; i += 32) s += wrow[i] * h[i];
  }
  s += __shfl_xor(s, 16);
  s += __shfl_xor(s, 8);
  s += __shfl_xor(s, 4);
  s += __shfl_xor(s, 2);
  s += __shfl_xor(s, 1);
  if (lane == 0) hproj[b * 16 + a] = s;
}

// ---------------------------------------------------------------------------
// K1: attention scores via WMMA (software-pipelined).
//   pre[m, a] = hproj[m%B, a] + enc_flat[m, :] @ Wa1p[a, :]^T   (m = s*B+b)
//   e[m] = sum_a tanh(pre[m,a]) * Wa2[a]
// grid = S*B/64 blocks, block = 128 (4 waves, 1 m-tile each)
// ---------------------------------------------------------------------------
__global__ __launch_bounds__(128) void attn_score_kernel(
    const float* __restrict__ enc, const float* __restrict__ Wa1p,
    const float* __restrict__ Wa2, const float* __restrict__ hproj,
    float* __restrict__ e) {
  const int lane = threadIdx.x & 31;
  const int wave = threadIdx.x >> 5;
  const int m0 = (blockIdx.x * 4 + wave) * 16;
  const int l16 = lane & 15;
  const int khalf = (lane >> 4) * 2;
  const int half8 = (lane >> 4) * 8;
  const float w2 = (l16 < CA) ? Wa2[l16] : 0.f;

  v8f acc;
#pragma unroll
  for (int r = 0; r < 8; ++r) {
    const int m = m0 + r + half8;
    const int b = m & (CB - 1);
    acc[r] = (l16 < CA) ? hproj[b * 16 + l16] : 0.f;
  }

  const float* Ap = enc + (long)(m0 + l16) * CH + khalf;
  const float* Bp = Wa1p + (long)l16 * CH + khalf;

  // software pipeline: loads for step k+4 issued before WMMA consumes step k
  v2f bv = *(const v2f*)(Bp);
  v2f av = *(const v2f*)(Ap);
#pragma unroll 4
  for (int k = 4; k < CH; k += 4) {
    v2f bn = *(const v2f*)(Bp + k);
    v2f an = *(const v2f*)(Ap + k);
    acc = wmma_f32_k4(av, bv, acc);
    bv = bn;
    av = an;
  }
  acc = wmma_f32_k4(av, bv, acc);

#pragma unroll
  for (int r = 0; r < 8; ++r) {
    float val = tanhf(acc[r]) * w2;
    val += __shfl_xor(val, 1);
    val += __shfl_xor(val, 2);
    val += __shfl_xor(val, 4);
    val += __shfl_xor(val, 8);
    if (l16 == 0) e[m0 + r + half8] = val;
  }
}

// ---------------------------------------------------------------------------
// K2: softmax over S for each b (in place: e -> alpha). grid = B, block = S
// ---------------------------------------------------------------------------
__global__ __launch_bounds__(512) void softmax_kernel(float* __restrict__ e) {
  const int b = blockIdx.x;
  const int s = threadIdx.x;
  __shared__ float red[CS];
  float v = e[s * CB + b];
  red[s] = v;
  __syncthreads();
  for (int off = CS / 2; off > 0; off >>= 1) {
    if (s < off) red[s] = fmaxf(red[s], red[s + off]);
    __syncthreads();
  }
  const float mx = red[0];
  __syncthreads();
  const float ev = expf(v - mx);
  red[s] = ev;
  __syncthreads();
  for (int off = CS / 2; off > 0; off >>= 1) {
    if (s < off) red[s] += red[s + off];
    __syncthreads();
  }
  e[s * CB + b] = ev / red[0];
}

// ---------------------------------------------------------------------------
// K3: xc[b, 0:E] = emb[dec_input[b], :];  xc[b, E+h] = sum_s alpha[s,b]*enc[s,b,h]
// grid = ((E+H)/256, B), block = 256
// ---------------------------------------------------------------------------
__global__ __launch_bounds__(256) void ctx_gather_kernel(
    const float* __restrict__ alpha, const float* __restrict__ enc,
    const float* __restrict__ emb, const int* __restrict__ dec_input,
    float* __restrict__ xc) {
  const int b = blockIdx.y;
  const int col = blockIdx.x * 256 + threadIdx.x;
  if (col < CE) {
    const int row = dec_input[b];
    xc[b * (CE + CH) + col] = emb[(long)row * CE + col];
  } else {
    const int h = col - CE;
    const float* ap = alpha + b;
    const float* ep = enc + (long)b * CH + h;
    float sum = 0.f;
#pragma unroll 4
    for (int s = 0; s < CS; ++s)
      sum += ap[s * CB] * ep[(long)s * (CB * CH)];
    xc[b * (CE + CH) + col] = sum;
  }
}

// ---------------------------------------------------------------------------
// K4/K6: generic out(64 x N) = X(64 x K) @ W(N x K)^T + bias(N) via WMMA f32,
// software-pipelined. block = 128 (4 waves). Each wave: one 16-col N tile and
// all four 16-row M tiles (B operand reused 4x per k-step). grid.x = N/64.
// ---------------------------------------------------------------------------
__global__ __launch_bounds__(128) void gemm_xt_kernel(
    const float* __restrict__ X, const float* __restrict__ W,
    const float* __restrict__ bias, float* __restrict__ out,
    int N, int K) {
  const int lane = threadIdx.x & 31;
  const int wave = threadIdx.x >> 5;
  const int n0 = (blockIdx.x * 4 + wave) * 16;
  const int l16 = lane & 15;
  const int khalf = (lane >> 4) * 2;

  v8f acc0 = {}, acc1 = {}, acc2 = {}, acc3 = {};
  const float* Wp = W + (long)(n0 + l16) * K + khalf;
  const float* X0 = X + (long)(0 * 16 + l16) * K + khalf;
  const float* X1 = X + (long)(1 * 16 + l16) * K + khalf;
  const float* X2 = X + (long)(2 * 16 + l16) * K + khalf;
  const float* X3 = X + (long)(3 * 16 + l16) * K + khalf;

  // software pipeline: loads for step k+4 issued before WMMAs consume step k
  v2f bv = *(const v2f*)(Wp);
  v2f a0 = *(const v2f*)(X0);
  v2f a1 = *(const v2f*)(X1);
  v2f a2 = *(const v2f*)(X2);
  v2f a3 = *(const v2f*)(X3);
#pragma unroll 4
  for (int k = 4; k < K; k += 4) {
    v2f bn  = *(const v2f*)(Wp + k);
    v2f a0n = *(const v2f*)(X0 + k);
    v2f a1n = *(const v2f*)(X1 + k);
    v2f a2n = *(const v2f*)(X2 + k);
    v2f a3n = *(const v2f*)(X3 + k);
    acc0 = wmma_f32_k4(a0, bv, acc0);
    acc1 = wmma_f32_k4(a1, bv, acc1);
    acc2 = wmma_f32_k4(a2, bv, acc2);
    acc3 = wmma_f32_k4(a3, bv, acc3);
    bv = bn; a0 = a0n; a1 = a1n; a2 = a2n; a3 = a3n;
  }
  acc0 = wmma_f32_k4(a0, bv, acc0);
  acc1 = wmma_f32_k4(a1, bv, acc1);
  acc2 = wmma_f32_k4(a2, bv, acc2);
  acc3 = wmma_f32_k4(a3, bv, acc3);

  const float bcol = bias[n0 + l16];
  const int half8 = (lane >> 4) * 8;
#pragma unroll
  for (int r = 0; r < 8; ++r) {
    const int rr = r + half8;
    out[(long)(0 * 16 + rr) * N + n0 + l16] = acc0[r] + bcol;
    out[(long)(1 * 16 + rr) * N + n0 + l16] = acc1[r] + bcol;
    out[(long)(2 * 16 + rr) * N + n0 + l16] = acc2[r] + bcol;
    out[(long)(3 * 16 + rr) * N + n0 + l16] = acc3[r] + bcol;
  }
}

// ---------------------------------------------------------------------------
// K5: GRU elementwise. grid = (H/256, B), block = 256
// ---------------------------------------------------------------------------
__global__ __launch_bounds__(256) void gru_kernel(
    const float* __restrict__ gi, const float* __restrict__ gh,
    const float* __restrict__ hlast, float* __restrict__ hnew) {
  const int b = blockIdx.y;
  const int h = blockIdx.x * 256 + threadIdx.x;
  const float* gib = gi + b * (3 * CH);
  const float* ghb = gh + b * (3 * CH);
  const float ir = gib[h], iz = gib[CH + h], in_ = gib[2 * CH + h];
  const float hr = ghb[h], hz = ghb[CH + h], hn = ghb[2 * CH + h];
  const float r = 1.f / (1.f + expf(-(ir + hr)));
  const float z = 1.f / (1.f + expf(-(iz + hz)));
  const float n = tanhf(in_ + r * hn);
  const float hl = hlast[b * CH + h];
  hnew[b * CH + h] = (1.f - z) * n + z * hl;
}

// ---------------------------------------------------------------------------
extern "C" void kernel_launch(void* const* d_in, const int* in_sizes, int n_in,
                              void* d_out, int out_size, void* d_ws, size_t ws_size,
                              hipStream_t stream) {
  const int*   dec_input = (const int*)  d_in[0];
  const float* dec_hidden= (const float*)d_in[1];   // (1,B,H) == h_last (B,H)
  const float* enc       = (const float*)d_in[2];   // (S,B,H)
  const float* emb       = (const float*)d_in[3];   // (V,E)
  const float* Wa1       = (const float*)d_in[4];   // (A, 2H)
  const float* Wa2       = (const float*)d_in[5];   // (1, A)
  const float* W_ih      = (const float*)d_in[6];   // (3H, E+H)
  const float* W_hh      = (const float*)d_in[7];   // (3H, H)
  const float* b_ih      = (const float*)d_in[8];
  const float* b_hh      = (const float*)d_in[9];
  const float* W_out     = (const float*)d_in[10];  // (V, H)
  const float* b_out     = (const float*)d_in[11];

  float* ws    = (float*)d_ws;
  float* hproj = ws;                          // B*16          = 1024
  float* Wa1p  = ws + 1024;                   // 16*H          = 16384
  float* e     = Wa1p + 16384;                // S*B           = 32768
  float* xc    = e + 32768;                   // B*(E+H)       = 131072
  float* gi    = xc + 131072;                 // B*3H          = 196608
  float* gh    = gi + 196608;                 // B*3H          = 196608

  float* logits = (float*)d_out;                       // (B, V)
  float* hnew   = (float*)d_out + (size_t)CB * CV;     // (B, H)

  // 0) zero-padded attention weight slab (16 x H)
  pad_wa1_kernel<<<dim3((16 * CH) / 256), 256, 0, stream>>>(Wa1, Wa1p);

  // 1) hproj = Wa1[:, :H] @ h_last^T  (per b)
  hproj_kernel<<<dim3(16, CB), 32, 0, stream>>>(Wa1, dec_hidden, hproj);

  // 2) attention scores e (WMMA) : S*B rows / 64 per block
  attn_score_kernel<<<dim3((CS * CB) / 64), 128, 0, stream>>>(enc, Wa1p, Wa2, hproj, e);

  // 3) softmax over S per b -> alpha (in place)
  softmax_kernel<<<dim3(CB), CS, 0, stream>>>(e);

  // 4) context + embedding gather -> xc
  ctx_gather_kernel<<<dim3((CE + CH) / 256, CB), 256, 0, stream>>>(e, enc, emb, dec_input, xc);

  // 5) gi = xc @ W_ih^T + b_ih ; gh = h_last @ W_hh^T + b_hh  (WMMA)
  gemm_xt_kernel<<<dim3((3 * CH) / 64), 128, 0, stream>>>(xc, W_ih, b_ih, gi, 3 * CH, CE + CH);
  gemm_xt_kernel<<<dim3((3 * CH) / 64), 128, 0, stream>>>(dec_hidden, W_hh, b_hh, gh, 3 * CH, CH);

  // 6) GRU gates -> h_new (written directly into output tail)
  gru_kernel<<<dim3(CH / 256, CB), 256, 0, stream>>>(gi, gh, dec_hidden, hnew);

  // 7) logits = h_new @ W_out^T + b_out  (WMMA)
  gemm_xt_kernel<<<dim3(CV / 64), 128, 0, stream>>>(hnew, W_out, b_out, logits, CV, CH);
}